// GATv2_10806137717385
// MI455X (gfx1250) — compile-verified
//
#include <hip/hip_runtime.h>
#include <math.h>

typedef __attribute__((ext_vector_type(2))) float v2f;
typedef __attribute__((ext_vector_type(8))) float v8f;

#define DIN 128   // input / output feature width (H*D)
#define HH  8     // heads
#define DDIM 16   // channels per head

// ---------------------------------------------------------------------------
// Kernel 0: init. out = 0 (accumulated into by atomics), m = -inf, den = 0.
// ---------------------------------------------------------------------------
__global__ __launch_bounds__(256) void gat_init(float* __restrict__ out,
                                                float* __restrict__ m_,
                                                float* __restrict__ den,
                                                int N) {
  int i = blockIdx.x * blockDim.x + threadIdx.x;
  if (i < N * DIN) out[i] = 0.0f;
  if (i < N * HH) {
    m_[i]  = __int_as_float(0xff800000);  // -inf
    den[i] = 0.0f;
  }
}

// ---------------------------------------------------------------------------
// Kernel 1: fused Q/K GEMM (WMMA f32 16x16x4) + per-node logit projections.
//   Q  = x @ Wq + bq                  -> stored [N,128]
//   ql[n,h] = sum_d Q[n,h*16+d] * attn_w[d,h]
//   kl[n,h] = sum_d (x@Wk+bk)[n,h*16+d] * attn_w[d,h]   (K never stored)
// One wave32 per (16-row tile, head). Block = 8 waves = all 8 heads.
// ---------------------------------------------------------------------------
__global__ __launch_bounds__(256) void gat_qk_wmma(
    const float* __restrict__ x,
    const float* __restrict__ Wq, const float* __restrict__ bq,
    const float* __restrict__ Wk, const float* __restrict__ bk,
    const float* __restrict__ attn_w,
    float* __restrict__ Q, float* __restrict__ ql, float* __restrict__ kl,
    int N) {
  const int lane  = threadIdx.x & 31;
  const int wave  = threadIdx.x >> 5;  // head index h = N-tile index (0..7)
  const int hi    = lane >> 4;         // lane half (0/1)
  const int lm    = lane & 15;
  const int mbase = blockIdx.x * 16;

  // A fragment rows: both halves cover M = lm; halves differ in K (0,1 vs 2,3)
  int arow = mbase + lm;
  if (arow >= N) arow = N - 1;  // clamp (keeps EXEC all-ones for WMMA)
  const int bcol = wave * 16 + lm;  // B column within this head's 16 channels

  const float* xr = x + (size_t)arow * DIN;
  v8f cq = {};
  v8f ck = {};
#pragma unroll
  for (int k = 0; k < DIN; k += 4) {
    const int kk = k + 2 * hi;
    v2f a;
    a.x = xr[kk];
    a.y = xr[kk + 1];
    v2f bqv, bkv;
    bqv.x = Wq[(size_t)kk * DIN + bcol];
    bqv.y = Wq[(size_t)(kk + 1) * DIN + bcol];
    bkv.x = Wk[(size_t)kk * DIN + bcol];
    bkv.y = Wk[(size_t)(kk + 1) * DIN + bcol];
    cq = __builtin_amdgcn_wmma_f32_16x16x4_f32(false, a, false, bqv, (short)0,
                                               cq, false, false);
    ck = __builtin_amdgcn_wmma_f32_16x16x4_f32(false, a, false, bkv, (short)0,
                                               ck, false, false);
  }

  const float biasq = bq[bcol];
  const float biask = bk[bcol];
  // attn_w is [D=16, H=8] row-major; this lane's column is d = lm, head = wave
  const float wsc = attn_w[lm * HH + wave];

  float qv[8], pq[8], pk[8];
#pragma unroll
  for (int v = 0; v < 8; ++v) {
    qv[v]     = cq[v] + biasq;
    float kvv = ck[v] + biask;
    pq[v]     = qv[v] * wsc;
    pk[v]     = kvv * wsc;
  }
  // Reduce the weighted column contributions across the 16 lanes of each half
  // (xor masks 1..8 stay within a half). After this every lane in a half holds
  // the full per-row dot products for its 8 rows.
#pragma unroll
  for (int mask = 1; mask <= 8; mask <<= 1) {
#pragma unroll
    for (int v = 0; v < 8; ++v) {
      pq[v] += __shfl_xor(pq[v], mask, 32);
      pk[v] += __shfl_xor(pk[v], mask, 32);
    }
  }

  // C/D layout: lane half 0 -> rows v, half 1 -> rows 8+v; col = lm.
#pragma unroll
  for (int v = 0; v < 8; ++v) {
    int row = mbase + 8 * hi + v;
    if (row < N) Q[(size_t)row * DIN + bcol] = qv[v];
  }
  if (lm == 0) {
#pragma unroll
    for (int v = 0; v < 8; ++v) {
      int row = mbase + 8 * hi + v;
      if (row < N) {
        ql[(size_t)row * HH + wave] = pq[v];
        kl[(size_t)row * HH + wave] = pk[v];
      }
    }
  }
}

// ---------------------------------------------------------------------------
// Float atomic-max via int/uint trick (valid with -inf initialization).
// ---------------------------------------------------------------------------
__device__ __forceinline__ void atomicMaxF32(float* addr, float val) {
  if (val >= 0.0f)
    atomicMax((int*)addr, __float_as_int(val));
  else
    atomicMin((unsigned int*)addr, __float_as_uint(val));
}

// ---------------------------------------------------------------------------
// Kernel 2: per-edge segment max of logits into m[dst,h].
// ---------------------------------------------------------------------------
__global__ __launch_bounds__(256) void gat_edge_max(
    const int* __restrict__ src, const int* __restrict__ dst,
    const float* __restrict__ ql, const float* __restrict__ kl,
    float* __restrict__ m_, int E) {
  int e = blockIdx.x * blockDim.x + threadIdx.x;
  if (e >= E) return;
  const int s = src[e], d = dst[e];
  const float* qs = ql + (size_t)s * HH;
  const float* kd = kl + (size_t)d * HH;
  float* md = m_ + (size_t)d * HH;
#pragma unroll
  for (int h = 0; h < HH; ++h) atomicMaxF32(&md[h], qs[h] + kd[h]);
}

// ---------------------------------------------------------------------------
// Kernel 3: per-edge exp / segment-sum denominator into den[dst,h].
// ---------------------------------------------------------------------------
__global__ __launch_bounds__(256) void gat_edge_den(
    const int* __restrict__ src, const int* __restrict__ dst,
    const float* __restrict__ ql, const float* __restrict__ kl,
    const float* __restrict__ m_, float* __restrict__ den, int E) {
  int e = blockIdx.x * blockDim.x + threadIdx.x;
  if (e >= E) return;
  const int s = src[e], d = dst[e];
  const float* qs = ql + (size_t)s * HH;
  const float* kd = kl + (size_t)d * HH;
  const float* md = m_ + (size_t)d * HH;
  float* dd = den + (size_t)d * HH;
#pragma unroll
  for (int h = 0; h < HH; ++h)
    atomicAdd(&dd[h], __expf(qs[h] + kd[h] - md[h]));
}

// ---------------------------------------------------------------------------
// Kernel 4: message scatter. One wave32 per edge; lane owns 4 channels.
//   attn[e,h] = exp(logit - m[dst,h]) / den[dst,h]
//   out[dst, c] += Q[src, c] * attn[e, c/16]
// Q and out both fit in the 192MB L2 -> gathers/atomics are L2-resident.
// ---------------------------------------------------------------------------
__global__ __launch_bounds__(256) void gat_edge_msg(
    const int* __restrict__ src, const int* __restrict__ dst,
    const float* __restrict__ Q, const float* __restrict__ ql,
    const float* __restrict__ kl, const float* __restrict__ m_,
    const float* __restrict__ den, float* __restrict__ out, int E) {
  int gid  = blockIdx.x * blockDim.x + threadIdx.x;
  int e    = gid >> 5;
  int lane = gid & 31;
  if (e >= E) return;
  const int s = src[e], d = dst[e];
  const int h = lane >> 2;  // 4 lanes per head
  const float lg   = ql[(size_t)s * HH + h] + kl[(size_t)d * HH + h];
  const float attn = __expf(lg - m_[(size_t)d * HH + h]) / den[(size_t)d * HH + h];
  const int c = lane * 4;
  const float4 qv = *(const float4*)(Q + (size_t)s * DIN + c);
  float* o = out + (size_t)d * DIN + c;
  atomicAdd(o + 0, qv.x * attn);
  atomicAdd(o + 1, qv.y * attn);
  atomicAdd(o + 2, qv.z * attn);
  atomicAdd(o + 3, qv.w * attn);
}

// ---------------------------------------------------------------------------
// Kernel 5: in-place ReLU on the pooled output.
// ---------------------------------------------------------------------------
__global__ __launch_bounds__(256) void gat_relu(float* __restrict__ out, int n) {
  int i = blockIdx.x * blockDim.x + threadIdx.x;
  if (i < n) out[i] = fmaxf(out[i], 0.0f);
}

// ---------------------------------------------------------------------------
extern "C" void kernel_launch(void* const* d_in, const int* in_sizes, int n_in,
                              void* d_out, int out_size, void* d_ws,
                              size_t ws_size, hipStream_t stream) {
  const float* x      = (const float*)d_in[0];
  const float* Wq     = (const float*)d_in[1];
  const float* bq     = (const float*)d_in[2];
  const float* Wk     = (const float*)d_in[3];
  const float* bk     = (const float*)d_in[4];
  const float* attn_w = (const float*)d_in[5];
  const int*   src    = (const int*)d_in[6];
  const int*   dst    = (const int*)d_in[7];

  const int N = in_sizes[0] / DIN;
  const int E = in_sizes[6];

  float* out = (float*)d_out;

  // Workspace layout (floats): Q[N*128] | ql[N*8] | kl[N*8] | m[N*8] | den[N*8]
  float* ws  = (float*)d_ws;
  float* Q   = ws;
  float* ql  = Q + (size_t)N * DIN;
  float* kl  = ql + (size_t)N * HH;
  float* m_  = kl + (size_t)N * HH;
  float* den = m_ + (size_t)N * HH;

  const int nOut = N * DIN;

  gat_init<<<(nOut + 255) / 256, 256, 0, stream>>>(out, m_, den, N);

  gat_qk_wmma<<<(N + 15) / 16, 256, 0, stream>>>(x, Wq, bq, Wk, bk, attn_w, Q,
                                                 ql, kl, N);

  gat_edge_max<<<(E + 255) / 256, 256, 0, stream>>>(src, dst, ql, kl, m_, E);

  gat_edge_den<<<(E + 255) / 256, 256, 0, stream>>>(src, dst, ql, kl, m_, den,
                                                    E);

  const long long msgThreads = (long long)E * 32;
  gat_edge_msg<<<(unsigned)((msgThreads + 255) / 256), 256, 0, stream>>>(
      src, dst, Q, ql, kl, m_, den, out, E);

  gat_relu<<<(nOut + 255) / 256, 256, 0, stream>>>(out, nOut);
}